// GraphSAGEBaseline_47648367182182
// MI455X (gfx1250) — compile-verified
//
#include <hip/hip_runtime.h>
#include <hip/hip_bf16.h>
#include <math.h>

typedef float v2f __attribute__((ext_vector_type(2)));
typedef float v8f __attribute__((ext_vector_type(8)));

#define N_CH 128
#define LDS_STRIDE 260          // 256 + 4 pad: 260 % 64 == 4 -> conflict-free row reads
#define ROWS_PER_BLOCK 32       // 2 row-tiles of 16; LDS = 32*260*4 = 33.3 KB

// ---------------- degree: deg[dst] += 1 ----------------
__global__ void __launch_bounds__(256) deg_kernel(const long long* __restrict__ dst,
                                                  float* __restrict__ deg, int n_edges) {
  int e = blockIdx.x * blockDim.x + threadIdx.x;
  if (e < n_edges) atomicAdd(&deg[(int)dst[e]], 1.0f);
}

// ---------------- scatter-sum: agg[dst] += feat[src], 128 ch, 32 lanes/edge ----------------
__global__ void __launch_bounds__(256) scatter_kernel(const float* __restrict__ feat,
                                                      float* __restrict__ agg,
                                                      const long long* __restrict__ src,
                                                      const long long* __restrict__ dst,
                                                      int n_edges) {
  long long gid = (long long)blockIdx.x * blockDim.x + threadIdx.x;
  int e = (int)(gid >> 5);
  int lane = (int)(gid & 31);
  if (e >= n_edges) return;
  int s = (int)src[e];
  int d = (int)dst[e];
  const float4 v = *(const float4*)(feat + (long long)s * N_CH + lane * 4);
  float* ap = agg + (long long)d * N_CH + lane * 4;
  atomicAdd(ap + 0, v.x);
  atomicAdd(ap + 1, v.y);
  atomicAdd(ap + 2, v.z);
  atomicAdd(ap + 3, v.w);
}

// Wcat(k,n) = k<128 ? Wl[k][n] : Wr[k-128][n]
__device__ __forceinline__ float wsel(const float* Wl, const float* Wr, int k, int n, int ldn) {
  return (k < 128) ? Wl[k * ldn + n] : Wr[(k - 128) * ldn + n];
}

// ---------------- layer 1 GEMM: H = relu([agg/deg | x] @ [W1l;W1r] + b1), N=128 ----------------
__global__ void __launch_bounds__(256) sage_gemm1_kernel(
    const float* __restrict__ agg, const float* __restrict__ deg,
    const float* __restrict__ feat,
    const float* __restrict__ Wl, const float* __restrict__ Wr,
    const float* __restrict__ bias, float* __restrict__ out, int n_nodes) {
  __shared__ float lds[ROWS_PER_BLOCK * LDS_STRIDE];
  const int tid = threadIdx.x;
  const int row_base = blockIdx.x * ROWS_PER_BLOCK;

  // Stage A-tile: 32 rows x 256 cols, column == tid -> fully coalesced global reads.
  {
    const int c = tid;
    for (int r = 0; r < ROWS_PER_BLOCK; ++r) {
      int gr = min(row_base + r, n_nodes - 1);
      float v;
      if (c < N_CH) {
        v = agg[(long long)gr * N_CH + c] / fmaxf(deg[gr], 1.0f);
      } else {
        v = feat[(long long)gr * N_CH + (c - N_CH)];
      }
      lds[r * LDS_STRIDE + c] = v;
    }
  }
  __syncthreads();

  const int wave = tid >> 5;
  const int lane = tid & 31;
  const int l15 = lane & 15;
  const int hi = lane >> 4;
  const int n = wave * 16 + l15;   // output column 0..127

  v8f acc0 = {};
  v8f acc1 = {};
  for (int k0 = 0; k0 < 256; k0 += 4) {
    const int kk = k0 + hi * 2;    // ISA: lanes 16-31 hold K=k0+2,k0+3
    v2f b;
    b.x = wsel(Wl, Wr, kk, n, 128);
    b.y = wsel(Wl, Wr, kk + 1, n, 128);
    const float* a0 = &lds[(0 * 16 + l15) * LDS_STRIDE + kk];
    const float* a1 = &lds[(1 * 16 + l15) * LDS_STRIDE + kk];
    v2f av0 = {a0[0], a0[1]};
    v2f av1 = {a1[0], a1[1]};
    acc0 = __builtin_amdgcn_wmma_f32_16x16x4_f32(false, av0, false, b, (short)0, acc0, false, false);
    acc1 = __builtin_amdgcn_wmma_f32_16x16x4_f32(false, av1, false, b, (short)0, acc1, false, false);
  }

  const float bn = bias[n];
  for (int r = 0; r < 8; ++r) {
    int m0 = row_base + 0 * 16 + r + hi * 8;   // C layout: VGPR r = row r (lanes 0-15) / r+8
    int m1 = row_base + 1 * 16 + r + hi * 8;
    if (m0 < n_nodes) out[(long long)m0 * N_CH + n] = fmaxf(acc0[r] + bn, 0.0f);
    if (m1 < n_nodes) out[(long long)m1 * N_CH + n] = fmaxf(acc1[r] + bn, 0.0f);
  }
}

// ---------------- layer 2 GEMM: logits = [agg/deg | h] @ [W2l;W2r] + b2, N=40 (pad 48) ----------------
__global__ void __launch_bounds__(256) sage_gemm2_kernel(
    const float* __restrict__ agg, const float* __restrict__ deg,
    const float* __restrict__ feat,
    const float* __restrict__ Wl, const float* __restrict__ Wr,
    const float* __restrict__ bias, float* __restrict__ out, int n_nodes) {
  __shared__ float lds[ROWS_PER_BLOCK * LDS_STRIDE];
  const int tid = threadIdx.x;
  const int row_base = blockIdx.x * ROWS_PER_BLOCK;

  {
    const int c = tid;
    for (int r = 0; r < ROWS_PER_BLOCK; ++r) {
      int gr = min(row_base + r, n_nodes - 1);
      float v;
      if (c < N_CH) {
        v = agg[(long long)gr * N_CH + c] / fmaxf(deg[gr], 1.0f);
      } else {
        v = feat[(long long)gr * N_CH + (c - N_CH)];
      }
      lds[r * LDS_STRIDE + c] = v;
    }
  }
  __syncthreads();

  const int wave = tid >> 5;
  if (wave < 6) {                 // wave-uniform: EXEC stays all-ones for WMMA
    const int rt = wave / 3;      // row-tile 0..1
    const int nt = wave % 3;      // n-tile 0..2 (cols 0..47, valid < 40)
    const int lane = tid & 31;
    const int l15 = lane & 15;
    const int hi = lane >> 4;
    const int n = nt * 16 + l15;
    const bool nok = (n < 40);

    v8f acc = {};
    for (int k0 = 0; k0 < 256; k0 += 4) {
      const int kk = k0 + hi * 2;
      v2f b;
      b.x = nok ? wsel(Wl, Wr, kk, n, 40) : 0.0f;
      b.y = nok ? wsel(Wl, Wr, kk + 1, n, 40) : 0.0f;
      const float* ap = &lds[(rt * 16 + l15) * LDS_STRIDE + kk];
      v2f a = {ap[0], ap[1]};
      acc = __builtin_amdgcn_wmma_f32_16x16x4_f32(false, a, false, b, (short)0, acc, false, false);
    }

    const float bn = nok ? bias[n] : 0.0f;
    for (int r = 0; r < 8; ++r) {
      int m = row_base + rt * 16 + r + hi * 8;
      if (m < n_nodes && nok) out[(long long)m * 40 + n] = acc[r] + bn;
    }
  }
}

// ---------------- row-wise log-softmax over 40 cols, one wave32 per row, in place ----------------
__global__ void __launch_bounds__(256) logsoftmax_kernel(float* __restrict__ out, int n_nodes) {
  int row = blockIdx.x * 8 + (threadIdx.x >> 5);
  int lane = threadIdx.x & 31;
  if (row >= n_nodes) return;
  float* rp = out + (long long)row * 40;
  float v0 = rp[lane];
  float v1 = (lane < 8) ? rp[32 + lane] : -INFINITY;
  float m = fmaxf(v0, v1);
  for (int o = 16; o > 0; o >>= 1) m = fmaxf(m, __shfl_xor(m, o, 32));
  float s = expf(v0 - m) + ((lane < 8) ? expf(v1 - m) : 0.0f);
  for (int o = 16; o > 0; o >>= 1) s += __shfl_xor(s, o, 32);
  float lse = m + logf(s);
  rp[lane] = v0 - lse;
  if (lane < 8) rp[32 + lane] = v1 - lse;
}

extern "C" void kernel_launch(void* const* d_in, const int* in_sizes, int n_in,
                              void* d_out, int out_size, void* d_ws, size_t ws_size,
                              hipStream_t stream) {
  const float*     x   = (const float*)d_in[0];
  const long long* ei  = (const long long*)d_in[1];   // jnp.int64 edge_index [2, E]
  const float*     W1l = (const float*)d_in[2];
  const float*     W1r = (const float*)d_in[3];
  const float*     b1  = (const float*)d_in[4];
  const float*     W2l = (const float*)d_in[5];
  const float*     W2r = (const float*)d_in[6];
  const float*     b2  = (const float*)d_in[7];
  float* out = (float*)d_out;

  const int n_nodes = in_sizes[0] / N_CH;   // 100000
  const int n_edges = in_sizes[1] / 2;      // 1.6M
  const long long* src = ei;
  const long long* dst = ei + n_edges;

  // workspace: B1 (agg, reused for both layers) | H (hidden) | deg
  float* B1  = (float*)d_ws;
  float* H   = B1 + (size_t)n_nodes * N_CH;
  float* deg = H + (size_t)n_nodes * N_CH;

  const int tb = 256;
  const int gblocks = (n_nodes + ROWS_PER_BLOCK - 1) / ROWS_PER_BLOCK;
  const long long sthreads = (long long)n_edges * 32;
  const int sblocks = (int)((sthreads + tb - 1) / tb);

  hipMemsetAsync(deg, 0, sizeof(float) * (size_t)n_nodes, stream);
  hipMemsetAsync(B1, 0, sizeof(float) * (size_t)n_nodes * N_CH, stream);

  deg_kernel<<<(n_edges + tb - 1) / tb, tb, 0, stream>>>(dst, deg, n_edges);
  scatter_kernel<<<sblocks, tb, 0, stream>>>(x, B1, src, dst, n_edges);
  sage_gemm1_kernel<<<gblocks, tb, 0, stream>>>(B1, deg, x, W1l, W1r, b1, H, n_nodes);

  hipMemsetAsync(B1, 0, sizeof(float) * (size_t)n_nodes * N_CH, stream);
  scatter_kernel<<<sblocks, tb, 0, stream>>>(H, B1, src, dst, n_edges);
  sage_gemm2_kernel<<<gblocks, tb, 0, stream>>>(B1, deg, H, W2l, W2r, b2, out, n_nodes);

  logsoftmax_kernel<<<(n_nodes + 7) / 8, tb, 0, stream>>>(out, n_nodes);
}